// Geo_GCN_6511170421701
// MI455X (gfx1250) — compile-verified
//
#include <hip/hip_runtime.h>
#include <hip/hip_bf16.h>

typedef float v2f __attribute__((ext_vector_type(2)));
typedef float v8f __attribute__((ext_vector_type(8)));

#define CIN  128
#define COUT 128
#define KSTEPS (CIN / 4)            // 32 k-steps of the 16x16x4 f32 WMMA
#define NTILES_O (COUT / 16)        // 8 output-column tiles
// interleaved fragment store: [ks][t][mat][lane] v2f
// per-ks chunk = 8 t * 2 mat * 32 lanes = 512 v2f = 4 KB (16-bit DS offsets OK)
#define FRAGS_TOTAL (KSTEPS * NTILES_O * 2 * 32)   // 16384 v2f = 128 KB

// ---------------------------------------------------------------------------
// K0: zero side_embed [n*CIN] f32 and deg [n] u32 in workspace
// ---------------------------------------------------------------------------
__global__ void gcn_zero_kernel(float* __restrict__ side,
                                unsigned int* __restrict__ deg,
                                int nc, int n) {
    int i = blockIdx.x * blockDim.x + threadIdx.x;
    if (i < nc) side[i] = 0.0f;
    if (i < n)  deg[i] = 0u;
}

// ---------------------------------------------------------------------------
// K1: deg[col[e]] += 1  (exact integer degree via global_atomic_add_u32)
// ---------------------------------------------------------------------------
__global__ void gcn_degree_kernel(const long long* __restrict__ ei,
                                  unsigned int* __restrict__ deg,
                                  int E) {
    int e = blockIdx.x * blockDim.x + threadIdx.x;
    if (e >= E) return;
    int c = (int)ei[(size_t)E + e];
    atomicAdd(&deg[c], 1u);
}

// ---------------------------------------------------------------------------
// K2: dinv[i] = deg>0 ? rsqrt(max(deg,1)) : 0
// ---------------------------------------------------------------------------
__global__ void gcn_dinv_kernel(const unsigned int* __restrict__ deg,
                                float* __restrict__ dinv, int n) {
    int i = blockIdx.x * blockDim.x + threadIdx.x;
    if (i >= n) return;
    unsigned int d = deg[i];
    float df = (float)d;
    dinv[i] = (d > 0u) ? __frsqrt_rn(fmaxf(df, 1.0f)) : 0.0f;
}

// ---------------------------------------------------------------------------
// K3: scatter: side[row] += exp(-d*d)*dinv[row]*dinv[col] * x[col]
// One wave32 per edge; lane L covers channels 4L..4L+3 with a float4 load of
// x[col] (L2-resident: x is 25.6MB << 192MB L2) + 4 global_atomic_add_f32.
// ---------------------------------------------------------------------------
__global__ void gcn_scatter_kernel(const float* __restrict__ x,
                                   const long long* __restrict__ ei,
                                   const float* __restrict__ dist,
                                   const float* __restrict__ dinv,
                                   float* __restrict__ side,
                                   int E) {
    int wavesPerBlock = blockDim.x >> 5;
    int e = blockIdx.x * wavesPerBlock + ((int)threadIdx.x >> 5);
    if (e >= E) return;                      // wave-uniform exit
    int lane = (int)threadIdx.x & 31;

    int r = (int)ei[e];
    int c = (int)ei[(size_t)E + e];
    float d = dist[e];
    float val = __expf(-d * d) * dinv[r] * dinv[c];

    const float4* xrow = (const float4*)(x + (size_t)c * CIN);
    float4 v = xrow[lane];
    float* dst = side + (size_t)r * CIN + lane * 4;
    __hip_atomic_fetch_add(dst + 0, val * v.x, __ATOMIC_RELAXED, __HIP_MEMORY_SCOPE_AGENT);
    __hip_atomic_fetch_add(dst + 1, val * v.y, __ATOMIC_RELAXED, __HIP_MEMORY_SCOPE_AGENT);
    __hip_atomic_fetch_add(dst + 2, val * v.z, __ATOMIC_RELAXED, __HIP_MEMORY_SCOPE_AGENT);
    __hip_atomic_fetch_add(dst + 3, val * v.w, __ATOMIC_RELAXED, __HIP_MEMORY_SCOPE_AGENT);
}

// ---------------------------------------------------------------------------
// K4: fused out = side@W0^T + (x*side)@W1^T via V_WMMA_F32_16X16X4_F32.
//
// Weight fragments pre-swizzled into LDS once per workgroup, INTERLEAVED as
// [ks][t][mat][lane] so all 16 fragments of a k-step sit in one 4 KB chunk:
// a single per-k-step base register + 16-bit DS immediate offsets covers
// every load (no per-load address VALU, no 64KB-offset overflow), and
// (b0,b1) pairs are 256 B apart -> ds_load_2addr_b64 pairing.
//
// Inner loop loads ALL 16 fragments into arrays first, then issues the 16
// WMMAs, so the DS clause is in flight while WMMAs drain with graduated
// s_wait_dscnt. 128 KB LDS -> 2 WGs/WGP (occupancy is LDS-capped);
// __launch_bounds__(256,1) lifts the VGPR budget so acc(64) + frags(32)
// stay in registers (verified spill-free at this pressure).
// ---------------------------------------------------------------------------
__global__ void __launch_bounds__(256, 1)
gcn_gemm_kernel(const float* __restrict__ x,
                const float* __restrict__ side,
                const float* __restrict__ W0,
                const float* __restrict__ W1,
                float* __restrict__ out,
                int n, int nTiles) {
    extern __shared__ char smem_raw[];
    v2f* ldsB = (v2f*)smem_raw;                  // [KSTEPS][NTILES_O][2][32]

    // ---- cooperative fill: swizzle W0/W1 into interleaved B-fragment layout
    for (int i = threadIdx.x; i < FRAGS_TOTAL; i += blockDim.x) {
        int flane = i & 31;
        int fmat  = (i >> 5) & 1;
        int ft    = (i >> 6) & 7;
        int fks   = i >> 9;
        int fhalf = flane >> 4;
        int fl    = flane & 15;
        int k0    = fks * 4 + fhalf * 2;         // B[k][n] = W[n][k]
        int ncol  = ft * 16 + fl;
        const float* p = (fmat ? W1 : W0) + (size_t)ncol * CIN + k0;
        v2f v; v[0] = p[0]; v[1] = p[1];
        ldsB[i] = v;
    }
    __syncthreads();

    int wavesPerBlock = blockDim.x >> 5;
    int tile = blockIdx.x * wavesPerBlock + ((int)threadIdx.x >> 5);
    if (tile >= nTiles) return;                  // wave-uniform: EXEC all-ones below
    int lane = (int)threadIdx.x & 31;
    int half = lane >> 4;                        // 0: k+0..1, 1: k+2..3
    int l    = lane & 15;

    int nodeA = tile * 16 + l;
    if (nodeA > n - 1) nodeA = n - 1;            // clamp loads (N=50000 tiles exactly)
    const float* sRow = side + (size_t)nodeA * CIN + half * 2;
    const float* xRow = x    + (size_t)nodeA * CIN + half * 2;

    v8f acc[NTILES_O];
#pragma unroll
    for (int t = 0; t < NTILES_O; ++t) acc[t] = (v8f)0.0f;

    for (int ks = 0; ks < KSTEPS; ++ks) {
        const float* sp = sRow + ks * 4;
        const float* xp = xRow + ks * 4;
        v2f aS; aS[0] = sp[0];         aS[1] = sp[1];          // side fragment
        v2f aB; aB[0] = xp[0] * aS[0]; aB[1] = xp[1] * aS[1];  // bi_embed fragment

        // one base covers the whole 4 KB k-step chunk via DS immediates
        const v2f* bp = ldsB + (size_t)ks * (NTILES_O * 2 * 32) + lane;
        v2f b0[NTILES_O], b1[NTILES_O];
#pragma unroll
        for (int t = 0; t < NTILES_O; ++t) {     // issue all 16 ds_loads first
            b0[t] = bp[t * 64];                  // byte off t*512
            b1[t] = bp[t * 64 + 32];             // byte off t*512 + 256
        }
#pragma unroll
        for (int t = 0; t < NTILES_O; ++t) {     // then drain 16 WMMAs
            acc[t] = __builtin_amdgcn_wmma_f32_16x16x4_f32(
                false, aS, false, b0[t], (short)0, acc[t], false, false);
            acc[t] = __builtin_amdgcn_wmma_f32_16x16x4_f32(
                false, aB, false, b1[t], (short)0, acc[t], false, false);
        }
    }

    // store D: VGPR j = row j (lanes 0-15) / row j+8 (lanes 16-31), col = t*16+l
#pragma unroll
    for (int t = 0; t < NTILES_O; ++t) {
#pragma unroll
        for (int j = 0; j < 8; ++j) {
            int node = tile * 16 + j + half * 8;
            if (node < n)
                out[(size_t)node * COUT + t * 16 + l] = acc[t][j];
        }
    }
}

// ---------------------------------------------------------------------------
// host launcher
// ---------------------------------------------------------------------------
extern "C" void kernel_launch(void* const* d_in, const int* in_sizes, int n_in,
                              void* d_out, int out_size, void* d_ws, size_t ws_size,
                              hipStream_t stream) {
    const float*     x    = (const float*)d_in[0];
    const long long* ei   = (const long long*)d_in[1];   // int64 [2,E]
    // d_in[2] = adj_weight, unused by the forward pass
    const float*     dist = (const float*)d_in[3];
    const float*     W0   = (const float*)d_in[4];
    const float*     W1   = (const float*)d_in[5];
    float*           out  = (float*)d_out;

    const int NC = in_sizes[0];          // N * 128
    const int E  = in_sizes[3];
    const int n  = NC / CIN;

    // workspace layout: side [NC] f32 | deg [n] u32 | dinv [n] f32
    float*        side = (float*)d_ws;
    unsigned int* deg  = (unsigned int*)(side + (size_t)NC);
    float*        dinv = (float*)(deg + n);

    // K0: zero side + deg
    {
        int threads = 256;
        int blocks  = (NC + threads - 1) / threads;
        gcn_zero_kernel<<<blocks, threads, 0, stream>>>(side, deg, NC, n);
    }
    // K1: degree
    {
        int threads = 256;
        int blocks  = (E + threads - 1) / threads;
        gcn_degree_kernel<<<blocks, threads, 0, stream>>>(ei, deg, E);
    }
    // K2: dinv
    {
        int threads = 256;
        int blocks  = (n + threads - 1) / threads;
        gcn_dinv_kernel<<<blocks, threads, 0, stream>>>(deg, dinv, n);
    }
    // K3: scatter (8 waves/block -> 8 edges/block)
    {
        int threads = 256;
        int edgesPerBlock = threads / 32;
        int blocks = (E + edgesPerBlock - 1) / edgesPerBlock;
        gcn_scatter_kernel<<<blocks, threads, 0, stream>>>(x, ei, dist, dinv, side, E);
    }
    // K4: fused dual GEMM via WMMA f32 16x16x4, weights staged in LDS
    {
        int nTiles = (n + 15) / 16;
        int threads = 256;                        // 8 waves -> 128 nodes/block
        int tilesPerBlock = threads / 32;
        int blocks = (nTiles + tilesPerBlock - 1) / tilesPerBlock;
        size_t ldsBytes = (size_t)FRAGS_TOTAL * sizeof(v2f);   // 128 KB
        gcn_gemm_kernel<<<blocks, threads, ldsBytes, stream>>>(x, side, W0, W1, out, n, nTiles);
    }
}